// GaussianHMM_79104707657983
// MI455X (gfx1250) — compile-verified
//
#include <hip/hip_runtime.h>

// Problem constants (reference: T=262144, S=128, F=16, all fp32)
#define T_STEPS 262144
#define S_STATES 128
#define F_FEAT 16

// Workspace layout (float offsets). Total ~81KB.
#define AEXP_OFF 0        // exp(log_softmax(trans, axis=1)) : 128*128
#define BP_OFF   16384    // packed B matrix pairs: 16 * 256
#define C0_OFF   20480    // per-state constant : 128
#define LS_OFF   20608    // log_softmax(start) : 128

typedef float v2f __attribute__((ext_vector_type(2)));
typedef float v4f __attribute__((ext_vector_type(4)));
typedef float v8f __attribute__((ext_vector_type(8)));

// ---------------------------------------------------------------------------
// Prep: per-state tables. One block of 128 threads (thread == state).
// ---------------------------------------------------------------------------
__global__ void __launch_bounds__(128)
hmm_prep(const float* __restrict__ start, const float* __restrict__ trans,
         const float* __restrict__ means, const float* __restrict__ vars,
         float* __restrict__ ws)
{
  const int s = threadIdx.x;

  float invv[F_FEAT], mi[F_FEAT];
  float lognorm = 0.f, smi2 = 0.f;
#pragma unroll
  for (int f = 0; f < F_FEAT; ++f) {
    float v  = fmaxf(vars[s * F_FEAT + f], 1e-6f);   // clip(var, 1e-6)
    float iv = 1.0f / v;
    float mu = means[s * F_FEAT + f];
    invv[f] = iv;
    mi[f]   = mu * iv;
    lognorm += __logf(6.2831853071795864f * v);
    smi2    += mu * mu * iv;
  }
  // logB[t,s] = c0[s] + sum_k A[t,k]*B[k,s];  A = [x^2 | x]
  ws[C0_OFF + s] = -0.5f * lognorm - 0.5f * smi2;

  // B pairs: Bp[u][col] = (B[2u][col], B[2u+1][col]), col = state
#pragma unroll
  for (int u = 0; u < 8; ++u) {                      // k < 16: coeff of x^2
    ws[BP_OFF + u * 256 + 2 * s]     = -0.5f * invv[2 * u];
    ws[BP_OFF + u * 256 + 2 * s + 1] = -0.5f * invv[2 * u + 1];
  }
#pragma unroll
  for (int u = 8; u < 16; ++u) {                     // k >= 16: coeff of x
    ws[BP_OFF + u * 256 + 2 * s]     = mi[2 * u - 16];
    ws[BP_OFF + u * 256 + 2 * s + 1] = mi[2 * u - 15];
  }

  // log_softmax(start_prob)
  float m = -3.4e38f;
  for (int i = 0; i < S_STATES; ++i) m = fmaxf(m, start[i]);
  float sum = 0.f;
  for (int i = 0; i < S_STATES; ++i) sum += __expf(start[i] - m);
  ws[LS_OFF + s] = start[s] - m - __logf(sum);

  // Aexp[i][j] = exp(log_softmax(trans, axis=1))   (row i of trans)
  const float* tr = trans + s * S_STATES;
  float mr = -3.4e38f;
  for (int j = 0; j < S_STATES; ++j) mr = fmaxf(mr, tr[j]);
  float sr = 0.f;
  for (int j = 0; j < S_STATES; ++j) sr += __expf(tr[j] - mr);
  float isr = 1.0f / sr;
  for (int j = 0; j < S_STATES; ++j)
    ws[AEXP_OFF + s * S_STATES + j] = __expf(tr[j] - mr) * isr;
}

// ---------------------------------------------------------------------------
// Emissions via V_WMMA_F32_16X16X4_F32. Each wave: one 16(t) x 128(s) tile.
// Writes logB directly into d_out (scan overwrites it in place with logalpha).
// ---------------------------------------------------------------------------
__global__ void __launch_bounds__(256)
hmm_emis(const float* __restrict__ x, const float* __restrict__ ws,
         float* __restrict__ out)
{
  const int lane = threadIdx.x & 31;
  const int wv   = threadIdx.x >> 5;
  const int h16  = lane >> 4;        // half-wave: selects K pair {0,1} vs {2,3}
  const int lr   = lane & 15;        // M index (A) / N index (B,C,D)
  const int t0   = (blockIdx.x * 8 + wv) * 16;
  const int row  = t0 + lr;

  // A-matrix operands: 16x4 fp32 -> 2 VGPRs.
  // lanes 0-15: (M=lr, K=4kb+0 / 4kb+1); lanes 16-31: (M=lr, K=4kb+2 / 4kb+3)
  const float* xr = x + (size_t)row * F_FEAT + h16 * 2;
  v2f xv[4];
#pragma unroll
  for (int q = 0; q < 4; ++q) xv[q] = *(const v2f*)(xr + q * 4);
  v2f amat[8];
#pragma unroll
  for (int q = 0; q < 4; ++q) { amat[q] = xv[q] * xv[q]; amat[q + 4] = xv[q]; }

  const float* bp = ws + BP_OFF;
  const float* c0 = ws + C0_OFF;

#pragma unroll
  for (int nb = 0; nb < 8; ++nb) {
    const int col = nb * 16 + lr;
    const float c = c0[col];
    v8f acc = {c, c, c, c, c, c, c, c};
#pragma unroll
    for (int kb = 0; kb < 8; ++kb) {
      v2f b = *(const v2f*)(bp + (kb * 2 + h16) * 256 + col * 2);
      acc = __builtin_amdgcn_wmma_f32_16x16x4_f32(
          false, amat[kb], false, b, (short)0, acc, false, false);
    }
    // D layout: VGPR r -> M = r + 8*h16, N = lr
    float* o = out + (size_t)(t0 + h16 * 8) * S_STATES + col;
#pragma unroll
    for (int r = 0; r < 8; ++r) o[(size_t)r * S_STATES] = acc[r];
  }
}

// ---------------------------------------------------------------------------
// Sequential forward scan: 1 block, 512 threads (16 waves).
//   thread = (g = tid>>6 in 0..7, jp = tid&63): owns i-range g*16..g*16+15 and
//   states j = 2jp, 2jp+1. Aexp columns live in registers (16 x v2f).
//   Wave 0 finalizes each step (log, max-reduce, exp) -> only 2 barriers/step.
// ---------------------------------------------------------------------------
__global__ void __launch_bounds__(512)
hmm_scan(const float* __restrict__ ws, float* __restrict__ out)
{
  __shared__ __align__(16) float accum[S_STATES];   // s[j] partial sums
  __shared__ __align__(16) float pdup[2 * S_STATES];// p[i] duplicated pairs

  const int tid  = threadIdx.x;
  const int g    = tid >> 6;
  const int jp   = tid & 63;
  const int lane = tid & 31;

  v2f Areg[16];
#pragma unroll
  for (int c = 0; c < 16; ++c)
    Areg[c] = *(const v2f*)(ws + AEXP_OFF + (g * 16 + c) * S_STATES + 2 * jp);

  if (tid < S_STATES) accum[tid] = 0.f;

  float M  = 0.f;
  v4f   p4 = {0.f, 0.f, 0.f, 0.f};

  if (tid < 32) {  // t = 0: logalpha0 = log_start + logB[0]
    v4f b0 = *(const v4f*)(out + lane * 4);
    v4f ls = *(const v4f*)(ws + LS_OFF + lane * 4);
    v4f la = ls + b0;
    *(v4f*)(out + lane * 4) = la;
    float mx = fmaxf(fmaxf(la.x, la.y), fmaxf(la.z, la.w));
#pragma unroll
    for (int off = 16; off > 0; off >>= 1)
      mx = fmaxf(mx, __shfl_xor(mx, off, 32));
    M = mx;
    p4.x = __expf(la.x - M); p4.y = __expf(la.y - M);
    p4.z = __expf(la.z - M); p4.w = __expf(la.w - M);
    v4f d0 = {p4.x, p4.x, p4.y, p4.y};
    v4f d1 = {p4.z, p4.z, p4.w, p4.w};
    *(v4f*)(pdup + lane * 8)     = d0;
    *(v4f*)(pdup + lane * 8 + 4) = d1;
  }
  __syncthreads();

  for (int t = 1; t < T_STEPS; ++t) {
    // ---- stage 1 (all 512 threads): partial matvec with v_pk_fma_f32 ----
    v2f acc2 = {0.f, 0.f};
    const float* pd = pdup + g * 32;
#pragma unroll
    for (int c = 0; c < 16; ++c) {
      v2f pp = *(const v2f*)(pd + 2 * c);   // broadcast {p_i, p_i}
      asm("v_pk_fma_f32 %0, %1, %2, %0" : "+v"(acc2) : "v"(pp), "v"(Areg[c]));
    }
    atomicAdd(&accum[2 * jp],     acc2.x);  // ds_add_f32
    atomicAdd(&accum[2 * jp + 1], acc2.y);
    __syncthreads();

    // ---- stage 2 (wave 0 only): finalize step t ----
    if (tid < 32) {
      v4f s4 = *(const v4f*)(&accum[lane * 4]);
      *(v4f*)(&accum[lane * 4]) = (v4f){0.f, 0.f, 0.f, 0.f};
      const size_t base = (size_t)t * S_STATES + lane * 4;
      v4f b4 = *(const v4f*)(out + base);   // logB[t]
      if (t + 32 < T_STEPS)
        __builtin_prefetch(out + base + 32 * S_STATES, 0, 3);
      v4f la;
      la.x = __logf(s4.x) + M + b4.x;
      la.y = __logf(s4.y) + M + b4.y;
      la.z = __logf(s4.z) + M + b4.z;
      la.w = __logf(s4.w) + M + b4.w;
      *(v4f*)(out + base) = la;             // overwrite logB with logalpha
      float mx = fmaxf(fmaxf(la.x, la.y), fmaxf(la.z, la.w));
#pragma unroll
      for (int off = 16; off > 0; off >>= 1)
        mx = fmaxf(mx, __shfl_xor(mx, off, 32));
      M = mx;
      p4.x = __expf(la.x - M); p4.y = __expf(la.y - M);
      p4.z = __expf(la.z - M); p4.w = __expf(la.w - M);
      v4f d0 = {p4.x, p4.x, p4.y, p4.y};
      v4f d1 = {p4.z, p4.z, p4.w, p4.w};
      *(v4f*)(pdup + lane * 8)     = d0;
      *(v4f*)(pdup + lane * 8 + 4) = d1;
    }
    __syncthreads();
  }

  // log_likelihood = M + log(sum_j p[j])
  if (tid < 32) {
    float sp = p4.x + p4.y + p4.z + p4.w;
#pragma unroll
    for (int off = 16; off > 0; off >>= 1)
      sp += __shfl_xor(sp, off, 32);
    if (lane == 0) out[(size_t)T_STEPS * S_STATES] = M + __logf(sp);
  }
}

// ---------------------------------------------------------------------------
extern "C" void kernel_launch(void* const* d_in, const int* in_sizes, int n_in,
                              void* d_out, int out_size, void* d_ws, size_t ws_size,
                              hipStream_t stream) {
  const float* x     = (const float*)d_in[0];  // (T, F)
  const float* start = (const float*)d_in[1];  // (S,)
  const float* trans = (const float*)d_in[2];  // (S, S)
  const float* means = (const float*)d_in[3];  // (S, F)
  const float* vars  = (const float*)d_in[4];  // (S, F)
  float* out = (float*)d_out;                  // T*S logalpha + 1 loglik
  float* ws  = (float*)d_ws;

  hipLaunchKernelGGL(hmm_prep, dim3(1), dim3(128), 0, stream,
                     start, trans, means, vars, ws);
  hipLaunchKernelGGL(hmm_emis, dim3(T_STEPS / 128), dim3(256), 0, stream,
                     x, ws, out);
  hipLaunchKernelGGL(hmm_scan, dim3(1), dim3(512), 0, stream, ws, out);
}